// OpticsSimulator_43044162240548
// MI455X (gfx1250) — compile-verified
//
#include <hip/hip_runtime.h>
#include <hip/hip_bf16.h>
#include <math.h>

// Problem constants (match reference)
#define Bc 4
#define Hc 1024
#define Wc 1024
#define Cc 3
#define NF 21
#define PSF_H 65
#define PSF_W 65
#define GRID_H 9
#define GRID_W 9
#define RES_H 31
#define RES_W 31

#define NN (PSF_H * PSF_W * Cc)       // 12675 : flattened psf "N" dimension
#define NT ((NN + 15) / 16)           // 793 N tiles
#define NNPAD (NT * 16)               // 12688 : N padded to whole tiles
#define MPAD 96                       // 81 grid cells padded to 6 WMMA tiles
#define MT (MPAD / 16)                // 6
#define KPAD 24                       // 21 fields padded to 6 k-steps of 4
#define PATCH 114                     // round(1024/9)

typedef float v2f __attribute__((ext_vector_type(2)));
typedef float v8f __attribute__((ext_vector_type(8)));

// ---------------------------------------------------------------------------
// Catmull-Rom style cubic weight, a = -0.5 (matches _cubic_weight)
// ---------------------------------------------------------------------------
__device__ __forceinline__ float cubicw(float t) {
    float at = fabsf(t);
    float at2 = at * at;
    float at3 = at2 * at;
    float w1 = 1.5f * at3 - 2.5f * at2 + 1.0f;
    float w2 = -0.5f * at3 + 2.5f * at2 - 4.0f * at + 2.0f;
    return (at <= 1.0f) ? w1 : ((at < 2.0f) ? w2 : 0.0f);
}

// ---------------------------------------------------------------------------
// Kernel 1: distortion warp + bicubic resample + relative illumination.
// One thread per (y, x); 4x4 cubic weights shared across all 12 (b,c) channels.
// Bandwidth-bound gather; global_prefetch warms the tap rows.
// ---------------------------------------------------------------------------
__global__ void k_distort(const float* __restrict__ img,
                          const float* __restrict__ shifts,
                          const float* __restrict__ ri_tab,
                          const float* __restrict__ field_map,
                          float* __restrict__ out) {
    int idx = blockIdx.x * blockDim.x + threadIdx.x;
    if (idx >= Hc * Wc) return;
    int y = idx / Wc;
    int x = idx - y * Wc;

    const float x1f = 0.70710678118654752f;   // Wc / sqrt(Hc^2 + Wc^2)
    const float y0f = 0.70710678118654752f;   // Hc / diag

    float xf = (float)x * (2.0f * x1f / (float)(Wc - 1)) - x1f;
    float yf = (float)y * (-2.0f * y0f / (float)(Hc - 1)) + y0f;
    float r = sqrtf(xf * xf + yf * yf);

    // linear interp of distortion shift at r*(NF-1)
    float si = fminf(fmaxf(r * (float)(NF - 1), 0.0f), (float)(NF - 1));
    float flo = floorf(si);
    int lo = (int)flo;
    int hi = (int)ceilf(si);
    float frac = si - flo;
    float shift = shifts[lo] * (1.0f - frac) + shifts[hi] * frac;

    float inv_r = 1.0f / r;                    // r > 0 for all pixel centers
    float dxf = shift * (xf * inv_r);          // shift * cos(atan2(yf, xf))
    float dyf = shift * (yf * inv_r);
    float dx = dxf * ((float)(Wc - 1) / (2.0f * x1f));
    float dy = dyf * ((float)(Hc - 1) / (-2.0f * y0f));

    float wxp = (float)x + dx;
    float wyp = (float)y + dy;

    float fx0 = floorf(wxp);
    float fy0 = floorf(wyp);
    int x0 = (int)fx0;
    int y0 = (int)fy0;
    float fx = wxp - fx0;
    float fy = wyp - fy0;

    // Warm the cache lines of the 4 tap rows (global_prefetch_b8)
    {
        int xi0 = min(max(x0 - 1, 0), Wc - 1);
#pragma unroll
        for (int i = 0; i < 4; ++i) {
            int yi = min(max(y0 + i - 1, 0), Hc - 1);
            __builtin_prefetch(img + ((size_t)yi * Wc + xi0) * Cc, 0, 3);
        }
    }

    float wxs[4], wys[4];
#pragma unroll
    for (int j = 0; j < 4; ++j) {
        wxs[j] = cubicw(fx - (float)(j - 1));
        wys[j] = cubicw(fy - (float)(j - 1));
    }

    float acc[Bc][Cc];
#pragma unroll
    for (int b = 0; b < Bc; ++b)
#pragma unroll
        for (int c = 0; c < Cc; ++c) acc[b][c] = 0.0f;

#pragma unroll
    for (int i = 0; i < 4; ++i) {
        int yi = min(max(y0 + i - 1, 0), Hc - 1);
        float wy = wys[i];
#pragma unroll
        for (int j = 0; j < 4; ++j) {
            int xi = min(max(x0 + j - 1, 0), Wc - 1);
            float w = wy * wxs[j];
            size_t base = ((size_t)yi * Wc + xi) * Cc;
#pragma unroll
            for (int b = 0; b < Bc; ++b) {
                const float* p = img + (size_t)b * Hc * Wc * Cc + base;
                acc[b][0] += w * p[0];
                acc[b][1] += w * p[1];
                acc[b][2] += w * p[2];
            }
        }
    }

    // relative illumination: linear interp on field_map value
    float fmv = field_map[idx];
    float ri_i = fminf(fmaxf(fmv * (float)(NF - 1), 0.0f), (float)(NF - 1));
    float rlo = floorf(ri_i);
    int ilo = (int)rlo;
    int ihi = (int)ceilf(ri_i);
    float rfrac = ri_i - rlo;
    float ri = ri_tab[ilo] * (1.0f - rfrac) + ri_tab[ihi] * rfrac;

#pragma unroll
    for (int b = 0; b < Bc; ++b) {
        size_t o = (((size_t)b * Hc + y) * Wc + x) * Cc;
#pragma unroll
        for (int c = 0; c < Cc; ++c) out[o + c] = acc[b][c] * ri;
    }
}

// ---------------------------------------------------------------------------
// Kernel 2: per-grid-cell field histogram -> weights (81 x 21), stored padded
// to MPAD x KPAD (zero fill) for the WMMA GEMM. rintf == np.round (half-even).
// ---------------------------------------------------------------------------
__global__ void k_weights(const float* __restrict__ field_map,
                          float* __restrict__ Wp) {
    int g = blockIdx.x;              // 0..MPAD-1
    __shared__ int cnt[NF];
    if (threadIdx.x < NF) cnt[threadIdx.x] = 0;
    __syncthreads();

    if (g < GRID_H * GRID_W) {
        int gy = g / GRID_W;
        int gx = g - gy * GRID_W;
        int r0 = (int)rintf((float)gy * ((float)(Hc - PATCH) / (float)(GRID_H - 1)));
        int c0 = (int)rintf((float)gx * ((float)(Wc - PATCH) / (float)(GRID_W - 1)));
        for (int p = threadIdx.x; p < PATCH * PATCH; p += blockDim.x) {
            int pr = p / PATCH;
            int pc = p - pr * PATCH;
            float v = field_map[(size_t)(r0 + pr) * Wc + (c0 + pc)];
            int d = (int)rintf(v * (float)(NF - 1));
            d = min(max(d, 0), NF - 1);
            atomicAdd(&cnt[d], 1);
        }
    }
    __syncthreads();
    if (threadIdx.x < KPAD) {
        float val = 0.0f;
        if (g < GRID_H * GRID_W && threadIdx.x < NF)
            val = (float)cnt[threadIdx.x] * (1.0f / (float)(PATCH * PATCH));
        Wp[(size_t)g * KPAD + threadIdx.x] = val;
    }
}

// ---------------------------------------------------------------------------
// Kernel 3a: repack sampled_psfs (NF x NN) into zero-padded WMMA-B layout:
// the (k, k+1) pair each lane needs is contiguous -> one b64 load per frag.
//   Bpack[((k>>1) * NNPAD + n) * 2 + (k & 1)] = P[k][n]   (0 outside)
// ---------------------------------------------------------------------------
__global__ void k_pack_b(const float* __restrict__ P,    // NF x NN
                         float* __restrict__ Bpack) {    // (KPAD/2) x NNPAD x 2
    int idx = blockIdx.x * blockDim.x + threadIdx.x;
    if (idx >= KPAD * NNPAD) return;
    int k = idx / NNPAD;
    int n = idx - k * NNPAD;
    float v = (k < NF && n < NN) ? P[(size_t)k * NN + n] : 0.0f;
    Bpack[(((size_t)(k >> 1) * NNPAD + n) << 1) + (k & 1)] = v;
}

// ---------------------------------------------------------------------------
// Kernel 3b: interp_psfs = weights @ psfs via V_WMMA_F32_16X16X4_F32.
// Branch-free: A from padded Wp (b64), B from packed Bpack (b64),
// unconditional D stores into N-padded interp. One wave per 16x16 tile.
// A layout per ISA §7.12.2: lanes 0-15 hold K=0,1 ; lanes 16-31 hold K=2,3.
// ---------------------------------------------------------------------------
__global__ void k_einsum_wmma(const float* __restrict__ Wp,     // MPAD x KPAD
                              const float* __restrict__ Bpack,  // packed B
                              float* __restrict__ interp) {     // MPAD x NNPAD
    int wave = (blockIdx.x * blockDim.x + threadIdx.x) >> 5;
    if (wave >= MT * NT) return;         // wave-uniform exit
    int lane = threadIdx.x & 31;
    int half = lane >> 4;
    int l15 = lane & 15;

    int mt = wave % MT;
    int nt = wave / MT;
    int m = mt * 16 + l15;
    int n = nt * 16 + l15;

    const v2f* Arow = (const v2f*)(Wp + (size_t)m * KPAD);   // KPAD/2 pairs

    v8f acc = {0.0f, 0.0f, 0.0f, 0.0f, 0.0f, 0.0f, 0.0f, 0.0f};
#pragma unroll
    for (int kk = 0; kk < KPAD / 4; ++kk) {
        int kp = kk * 2 + half;                               // pair index k0/2
        v2f a = Arow[kp];
        v2f b = *(const v2f*)(Bpack + (((size_t)kp * NNPAD + n) << 1));
        acc = __builtin_amdgcn_wmma_f32_16x16x4_f32(
            /*neg_a=*/false, a, /*neg_b=*/false, b,
            /*c_mod=*/(short)0, acc, /*reuse_a=*/false, /*reuse_b=*/false);
    }

    // D layout: VGPR r -> row M = r + 8*half, col N = lane&15
    int mbase = mt * 16 + half * 8;
#pragma unroll
    for (int r = 0; r < 8; ++r)
        interp[(size_t)(mbase + r) * NNPAD + n] = acc[r];
}

// ---------------------------------------------------------------------------
// Kernel 4: rotate each interpolated PSF by -atan2(x_center, y_center),
// bilinear with zero-weight outside (matches _rotate_bilinear).
// ---------------------------------------------------------------------------
__global__ void k_rotate(const float* __restrict__ interp,   // MPAD x NNPAD
                         const float* __restrict__ x_map,
                         const float* __restrict__ y_map,
                         float* __restrict__ rot) {          // 81 x NN
    int g = blockIdx.y;
    int gy = g / GRID_W;
    int gx = g - gy * GRID_W;
    float x0m = x_map[0], x1m = x_map[Wc - 1];
    float y0m = y_map[0], y1m = y_map[Hc - 1];
    float xc = ((float)gx + 0.5f) / (float)GRID_W * (x1m - x0m) + x0m;
    float yc = ((float)gy + 0.5f) / (float)GRID_H * (y1m - y0m) + y0m;
    float ang = -atan2f(xc, yc);         // rotate by -angles[g]
    float ca = cosf(ang);
    float sa = sinf(ang);
    const float ctr = (float)(PSF_H - 1) * 0.5f;

    int p = blockIdx.x * blockDim.x + threadIdx.x;
    if (p >= PSF_H * PSF_W) return;
    int yy = p / PSF_W;
    int xx = p - yy * PSF_W;

    float sx = ca * ((float)xx - ctr) - sa * ((float)yy - ctr) + ctr;
    float sy = sa * ((float)xx - ctr) + ca * ((float)yy - ctr) + ctr;
    float fsx = floorf(sx), fsy = floorf(sy);
    int xs = (int)fsx, ys = (int)fsy;
    float fx = sx - fsx, fy = sy - fsy;

    float o0 = 0.0f, o1 = 0.0f, o2 = 0.0f;
#pragma unroll
    for (int di = 0; di < 2; ++di) {
#pragma unroll
        for (int dj = 0; dj < 2; ++dj) {
            int yi = ys + di, xi = xs + dj;
            float w = (di ? fy : 1.0f - fy) * (dj ? fx : 1.0f - fx);
            bool valid = (yi >= 0) && (yi < PSF_H) && (xi >= 0) && (xi < PSF_W);
            int yic = min(max(yi, 0), PSF_H - 1);
            int xic = min(max(xi, 0), PSF_W - 1);
            const float* t = interp + (size_t)g * NNPAD + ((size_t)yic * PSF_W + xic) * Cc;
            float wv = valid ? w : 0.0f;
            o0 += wv * t[0];
            o1 += wv * t[1];
            o2 += wv * t[2];
        }
    }
    float* o = rot + (size_t)g * NN + (size_t)p * Cc;
    o[0] = o0; o[1] = o1; o[2] = o2;
}

// ---------------------------------------------------------------------------
// Kernel 5: antialiased linear resize 65x65 -> 31x31 (triangle kernel,
// radius 1/scale, per-position weight renormalization == jax.image.resize),
// fused with per-(g, c) sum-normalization through LDS. One block per PSF.
// ---------------------------------------------------------------------------
__global__ void k_resize_norm(const float* __restrict__ rot,  // 81 x NN
                              float* __restrict__ outp) {     // 81 x 31*31*3
    int g = blockIdx.x;
    __shared__ float res[RES_H * RES_W * Cc];
    __shared__ float csum[Cc];
    if (threadIdx.x < Cc) csum[threadIdx.x] = 0.0f;
    __syncthreads();

    const float scale = (float)RES_H / (float)PSF_H;   // < 1 (downsample)
    const float radius = 1.0f / scale;

    float part0 = 0.0f, part1 = 0.0f, part2 = 0.0f;
    for (int p = threadIdx.x; p < RES_H * RES_W; p += blockDim.x) {
        int oy = p / RES_W;
        int ox = p - oy * RES_W;
        float cy = ((float)oy + 0.5f) / scale - 0.5f;
        float cx = ((float)ox + 0.5f) / scale - 0.5f;

        // x taps
        int xlo = max(0, (int)ceilf(cx - radius));
        int xhi = min(PSF_W - 1, (int)floorf(cx + radius));
        float wxv[8];
        int nx = xhi - xlo + 1;
        float wxsum = 0.0f;
        for (int j = 0; j < nx; ++j) {
            float w = fmaxf(0.0f, 1.0f - fabsf((float)(xlo + j) - cx) * scale);
            wxv[j] = w;
            wxsum += w;
        }
        // y taps
        int ylo = max(0, (int)ceilf(cy - radius));
        int yhi = min(PSF_H - 1, (int)floorf(cy + radius));
        float a0 = 0.0f, a1 = 0.0f, a2 = 0.0f;
        float wysum = 0.0f;
        for (int i = ylo; i <= yhi; ++i) {
            float wy = fmaxf(0.0f, 1.0f - fabsf((float)i - cy) * scale);
            wysum += wy;
            const float* row = rot + (size_t)g * NN + (size_t)i * PSF_W * Cc;
            for (int j = 0; j < nx; ++j) {
                float w = wy * wxv[j];
                const float* t = row + (size_t)(xlo + j) * Cc;
                a0 += w * t[0];
                a1 += w * t[1];
                a2 += w * t[2];
            }
        }
        float inv = 1.0f / (wxsum * wysum);
        float v0 = a0 * inv, v1 = a1 * inv, v2 = a2 * inv;
        res[p * Cc + 0] = v0;
        res[p * Cc + 1] = v1;
        res[p * Cc + 2] = v2;
        part0 += v0; part1 += v1; part2 += v2;
    }
    atomicAdd(&csum[0], part0);
    atomicAdd(&csum[1], part1);
    atomicAdd(&csum[2], part2);
    __syncthreads();

    float i0 = 1.0f / csum[0];
    float i1 = 1.0f / csum[1];
    float i2 = 1.0f / csum[2];
    for (int p = threadIdx.x; p < RES_H * RES_W; p += blockDim.x) {
        size_t o = (size_t)g * RES_H * RES_W * Cc + (size_t)p * Cc;
        outp[o + 0] = res[p * Cc + 0] * i0;
        outp[o + 1] = res[p * Cc + 1] * i1;
        outp[o + 2] = res[p * Cc + 2] * i2;
    }
}

// ---------------------------------------------------------------------------
extern "C" void kernel_launch(void* const* d_in, const int* in_sizes, int n_in,
                              void* d_out, int out_size, void* d_ws, size_t ws_size,
                              hipStream_t stream) {
    const float* img    = (const float*)d_in[0];  // (4,1024,1024,3)
    const float* psfs   = (const float*)d_in[1];  // (21,65,65,3)
    const float* shifts = (const float*)d_in[2];  // (21,)
    const float* ri     = (const float*)d_in[3];  // (21,)
    const float* fmap   = (const float*)d_in[4];  // (1024,1024)
    const float* xmap   = (const float*)d_in[5];  // (1024,)
    const float* ymap   = (const float*)d_in[6];  // (1024,)
    float* out = (float*)d_out;

    // workspace layout (floats)
    float* Wp     = (float*)d_ws;                        // MPAD*KPAD (<4096)
    float* Bpack  = Wp + 4096;                           // KPAD*NNPAD
    float* interp = Bpack + (size_t)KPAD * NNPAD;        // MPAD*NNPAD
    float* rot    = interp + (size_t)MPAD * NNPAD;       // 81*NN

    // 1) main image path: distortion + bicubic + illumination (bandwidth bound)
    k_distort<<<(Hc * Wc) / 256, 256, 0, stream>>>(img, shifts, ri, fmap, out);

    // 2) field histogram -> padded weight matrix
    k_weights<<<MPAD, 256, 0, stream>>>(fmap, Wp);

    // 3a) repack psfs into zero-padded WMMA-B fragment layout
    k_pack_b<<<(KPAD * NNPAD + 255) / 256, 256, 0, stream>>>(psfs, Bpack);

    // 3b) weights @ psfs GEMM on the WMMA pipe (branch-free inner loop)
    int waves = MT * NT;
    int blocks = (waves + 7) / 8;                        // 8 waves / 256-thread block
    k_einsum_wmma<<<blocks, 256, 0, stream>>>(Wp, Bpack, interp);

    // 4) per-cell rotation
    dim3 rgrid((PSF_H * PSF_W + 255) / 256, GRID_H * GRID_W);
    k_rotate<<<rgrid, 256, 0, stream>>>(interp, xmap, ymap, rot);

    // 5) antialiased resize + normalization, fused in LDS
    float* out_psf = out + (size_t)Bc * Hc * Wc * Cc;
    k_resize_norm<<<GRID_H * GRID_W, 256, 0, stream>>>(rot, out_psf);
}